// KNN_18872086298689
// MI455X (gfx1250) — compile-verified
//
#include <hip/hip_runtime.h>
#include <hip/hip_bf16.h>

// KNN on gfx1250: dist^2 = |q|^2 + |c|^2 - 2 q.c, with the q.c 16x16 tile
// computed by V_WMMA_F32_16X16X4_F32 (K=3 padded to 4). Selection (top-20)
// runs on the VALU, software-pipelined against the next tile's WMMA
// (XDL WMMA co-executes with VALU on CDNA5).

typedef float v2f __attribute__((ext_vector_type(2)));
typedef float v8f __attribute__((ext_vector_type(8)));

#define KNN_K 20
#define CTILE 1024              // candidates staged into LDS per round
#define NTILES (CTILE / 16)     // 16-candidate WMMA tiles per round
#define BLOCK_THREADS 256       // 8 waves (wave32)
#define WAVES_PER_BLOCK 8       // one 16-query tile per wave

// Shared memory:
//   staging region : CTILE * 4 floats (x,y,z,norm)            = 16 KB
//   merge region   : WAVES * 32 lanes * (20 dist + 20 idx)    = 40 KB
// One buffer sized for the larger use (reused after the scan loop).
#define SMEM_FLOATS (WAVES_PER_BLOCK * 32 * KNN_K * 2)   // 10240 floats = 40 KB

__global__ __launch_bounds__(BLOCK_THREADS)
void knn_wmma_16x16x4(const float* __restrict__ pts,   // [n,3] barycenters
                      float* __restrict__ out,         // [n,KNN_K] indices as f32
                      int n)
{
    __shared__ float sm[SMEM_FLOATS];

    const int lane = threadIdx.x & 31;
    const int wave = threadIdx.x >> 5;
    const int col  = lane & 15;          // query column within the 16-query tile
    const bool hi  = lane >= 16;         // upper half-wave: K=2..3 / rows 8..15

    // ---- load this lane's query (B operand) ----
    const int qtile = blockIdx.x * WAVES_PER_BLOCK + wave;
    const int q     = qtile * 16 + col;
    const float qx = pts[q * 3 + 0];
    const float qy = pts[q * 3 + 1];
    const float qz = pts[q * 3 + 2];
    const float qn = qx * qx + qy * qy + qz * qz;

    // B (4x16): lanes 0-15 hold (K=0,K=1) = (x,y); lanes 16-31 hold (K=2,K=3) = (z,0)
    v2f bmat;
    bmat.x = hi ? qz : qx;
    bmat.y = hi ? 0.0f : qy;

    // A-operand fetch: staged candidate c is (x,y,z,norm) at sm[tile + c*4].
    // Lane layout wants lo lanes -> (x,y) [K=0,1], hi lanes -> (z,0) [K=2,3].
    // One aligned b64 load at float offset col*4 + (hi?2:0), then mask y on hi
    // lanes (they load (z,norm); norm must become the K=3 zero pad).
    const int aoff = col * 4 + (hi ? 2 : 0);

    // ---- per-lane sorted top-K (ascending squared distance) ----
    float best_d[KNN_K];
    int   best_i[KNN_K];
#pragma unroll
    for (int i = 0; i < KNN_K; ++i) { best_d[i] = 3.4e38f; best_i[i] = 0; }

    const int mbase = hi ? 8 : 0;        // candidate rows this half-wave owns in D

    // fully-inlined helpers -------------------------------------------------
    auto load_a = [&](int ct) -> v2f {
        v2f a = *(const v2f*)(&sm[ct * 64 + aoff]);   // 8B-aligned ds_load_b64
        a.y = hi ? 0.0f : a.y;                        // single v_cndmask
        return a;
    };
    auto process = [&](const v8f& dacc, int ct, int cbase) {
        const float* tile = &sm[ct * 64];
        const int cand0 = cbase + ct * 16 + mbase;
#pragma unroll
        for (int r = 0; r < 8; ++r) {
            const float cn = tile[(mbase + r) * 4 + 3];
            const float d2 = fmaf(-2.0f, dacc[r], qn + cn);
            const int   ci = cand0 + r;
            if (d2 < best_d[KNN_K - 1]) {
                best_d[KNN_K - 1] = d2;
                best_i[KNN_K - 1] = ci;
                // one bubble pass restores sorted order (only last out of place)
#pragma unroll
                for (int j = KNN_K - 1; j > 0; --j) {
                    const float dj = best_d[j], djm = best_d[j - 1];
                    const int   ij = best_i[j], ijm = best_i[j - 1];
                    const bool  sw = dj < djm;
                    best_d[j]     = sw ? djm : dj;
                    best_d[j - 1] = sw ? dj  : djm;
                    best_i[j]     = sw ? ijm : ij;
                    best_i[j - 1] = sw ? ij  : ijm;
                }
            }
        }
    };
    // -----------------------------------------------------------------------

    // ---- scan all candidates in LDS-staged rounds ----
    for (int cbase = 0; cbase < n; cbase += CTILE) {
        // stage CTILE candidates as (x,y,z,|c|^2); 4 per thread
#pragma unroll
        for (int t = 0; t < CTILE / BLOCK_THREADS; ++t) {
            const int s = t * BLOCK_THREADS + threadIdx.x;
            const int c = cbase + s;
            const float x = pts[c * 3 + 0];
            const float y = pts[c * 3 + 1];
            const float z = pts[c * 3 + 2];
            sm[s * 4 + 0] = x;
            sm[s * 4 + 1] = y;
            sm[s * 4 + 2] = z;
            sm[s * 4 + 3] = x * x + y * y + z * z;
            // pull the next round toward L2/WGP$ while we compute this one
            const int cnext = c + CTILE;
            if (cnext < n) __builtin_prefetch(&pts[cnext * 3], 0, 1);
        }
        __syncthreads();

        // software pipeline: WMMA for tile ct+1 in flight while the VALU
        // selection network consumes tile ct.
        v2f a0 = load_a(0);
        v8f zc = {};
        v8f dcur = __builtin_amdgcn_wmma_f32_16x16x4_f32(
            false, a0, false, bmat, (short)0, zc, false, false);
        for (int ct = 0; ct < NTILES - 1; ++ct) {
            v2f an = load_a(ct + 1);
            v8f zn = {};
            v8f dnext = __builtin_amdgcn_wmma_f32_16x16x4_f32(
                false, an, false, bmat, (short)0, zn, false, false);
            process(dcur, ct, cbase);
            dcur = dnext;
        }
        process(dcur, NTILES - 1, cbase);

        __syncthreads();   // also protects LDS reuse (merge region) after the loop
    }

    // ---- merge the two half-wave lists for each query via LDS ----
    // Per-wave region: 32 lanes * (20 dists + 20 idx-as-float)
    float* wreg = &sm[wave * 32 * KNN_K * 2];
    float* mine = &wreg[lane * (KNN_K * 2)];
#pragma unroll
    for (int i = 0; i < KNN_K; ++i) {
        mine[i]         = best_d[i];
        mine[KNN_K + i] = (float)best_i[i];
    }
    // Same-wave LDS ops are in-order and lanes are lockstep: no barrier needed.
    if (!hi) {
        const float* a = &wreg[lane * (KNN_K * 2)];
        const float* b = &wreg[(lane + 16) * (KNN_K * 2)];
        float* o = &out[q * KNN_K];
        int p = 0, s = 0;
#pragma unroll
        for (int j = 0; j < KNN_K; ++j) {
            const float da = a[p], db = b[s];
            if (da <= db) { o[j] = a[KNN_K + p]; ++p; }
            else          { o[j] = b[KNN_K + s]; ++s; }
        }
    }
}

extern "C" void kernel_launch(void* const* d_in, const int* in_sizes, int n_in,
                              void* d_out, int out_size, void* d_ws, size_t ws_size,
                              hipStream_t stream) {
    (void)n_in; (void)out_size; (void)d_ws; (void)ws_size;
    // d_in[0] = x [N,3] (row count only), d_in[1] = barycenters [N,3],
    // d_in[2] = k (=20), d_in[3] = batch_size (unused: tiling is internal)
    const float* pts = (const float*)d_in[1];
    float* out = (float*)d_out;
    const int n = in_sizes[1] / 3;                 // 65536
    const int blocks = n / (16 * WAVES_PER_BLOCK); // 512 blocks, 8 waves each
    knn_wmma_16x16x4<<<blocks, BLOCK_THREADS, 0, stream>>>(pts, out, n);
}